// NuFFT_21775484191178
// MI455X (gfx1250) — compile-verified
//
#include <hip/hip_runtime.h>
#include <math.h>

typedef float v2f __attribute__((ext_vector_type(2)));
typedef float v8f __attribute__((ext_vector_type(8)));

#define NPIX   1024
#define NCHAN  4
#define NVIS   200000
#define GSZ    2048
#define JTAP   6
#define KBBETA 14.04f   // 2.34 * J

// SCALE = 1000 * DL * G, DL = 0.005 * pi/(180*3600)
constexpr double PI_D  = 3.14159265358979323846;
constexpr double DL_D  = 0.005 * PI_D / (180.0 * 3600.0);
constexpr float  UVSCL = (float)(1000.0 * DL_D * (double)GSZ);

// ---------------------------------------------------------------- KB kernel
__device__ __forceinline__ float i0f(float ax) {
    if (ax <= 3.75f) {
        float t = ax * (1.0f / 3.75f); t *= t;
        return 1.0f + t*(3.5156229f + t*(3.0899424f + t*(1.2067492f +
                     t*(0.2659732f + t*(0.0360768f + t*0.0045813f)))));
    } else {
        float t = 3.75f / ax;
        return expf(ax) * rsqrtf(ax) *
               (0.39894228f + t*(0.01328592f + t*(0.00225319f + t*(-0.00157565f +
                t*(0.00916281f + t*(-0.02057706f + t*(0.02635537f +
                t*(-0.01647633f + t*0.00392377f))))))));
    }
}
__device__ __forceinline__ float kbw(float d) {
    float q = d * (1.0f / 3.0f);          // 2*d/J with J=6
    float s = 1.0f - q * q;
    if (s <= 0.0f) return 0.0f;
    return i0f(KBBETA * sqrtf(s)) * (1.0f / 6.0f);
}

// ------------------------------------------------------------- apodization
__global__ void apod_kernel(float* apod) {
    int i = blockIdx.x * blockDim.x + threadIdx.x;
    if (i >= NPIX) return;
    float n   = (float)(i - NPIX / 2) / (float)GSZ;
    float pjn = (float)(PI_D * JTAP) * n;
    float arg = KBBETA * KBBETA - pjn * pjn;
    float sq  = sqrtf(fabsf(arg));
    float ft;
    if (arg > 0.0f)      ft = sinhf(sq) / sq;
    else if (sq < 1e-12f) ft = 1.0f;
    else                  ft = sinf(sq) / sq;
    apod[i] = 1.0f / ft;
}

// -------------------------------------------- pad + apodize + fftshift fill
__global__ void fill_grid(const float* __restrict__ img,
                          const float* __restrict__ apod,
                          float2* __restrict__ buf) {
    size_t idx = (size_t)blockIdx.x * blockDim.x + threadIdx.x; // c,y,x packed
    int    x   = (int)(idx & (GSZ - 1));
    size_t r   = idx >> 11;
    int    y   = (int)(r & (GSZ - 1));
    int    c   = (int)(r >> 11);
    int ys = (y + GSZ / 2) & (GSZ - 1);   // roll(-GSZ/2) inverse map
    int xs = (x + GSZ / 2) & (GSZ - 1);
    float2 v = {0.0f, 0.0f};
    if (ys < NPIX && xs < NPIX)
        v.x = img[((size_t)c * NPIX + ys) * NPIX + xs] * apod[ys] * apod[xs];
    buf[idx] = v;
}

// --------------------------------------------------- WMMA complex 16x16x16
__device__ __forceinline__ void wmma_cmul(const v2f Ar[4], const v2f Ai[4],
                                          const v2f AiN[4],
                                          const v2f Br[4], const v2f Bi[4],
                                          v8f& Dr, v8f& Di) {
#pragma unroll
    for (int kc = 0; kc < 4; ++kc) {
        Dr = __builtin_amdgcn_wmma_f32_16x16x4_f32(false, Ar[kc],  false, Br[kc], (short)0, Dr, false, false);
        Dr = __builtin_amdgcn_wmma_f32_16x16x4_f32(false, AiN[kc], false, Bi[kc], (short)0, Dr, false, false);
        Di = __builtin_amdgcn_wmma_f32_16x16x4_f32(false, Ar[kc],  false, Bi[kc], (short)0, Di, false, false);
        Di = __builtin_amdgcn_wmma_f32_16x16x4_f32(false, Ai[kc],  false, Br[kc], (short)0, Di, false, false);
    }
}

// ---------------------------------------------------------------- row FFTs
// 2048 = 8 * 16 * 16 Cooley-Tukey; DFT-8/16 stages as 16x16 WMMA matmuls.
// One 256-thread block (8 wave32) per row; one WMMA tile per wave per stage.
__global__ __launch_bounds__(256) void fft_rows(float2* __restrict__ buf) {
    __shared__ float sRe[GSZ], sIm[GSZ], tRe[GSZ], tIm[GSZ];
    const int tid  = threadIdx.x;
    const int lane = tid & 31;
    const int wave = tid >> 5;

    float2* row = buf + (size_t)blockIdx.x * GSZ;

#pragma unroll
    for (int q = 0; q < 8; ++q) {
        int e = tid + 256 * q;
        float2 v = row[e];
        sRe[e] = v.x; sIm[e] = v.y;
    }
    __syncthreads();

    const int m    = lane & 15;          // A-matrix output row
    const int ksel = (lane >> 4) << 1;   // A K-offset (0 or 2)
    const int col  = lane & 15;          // B/D column
    const int hi   = lane >> 4;          // B row-select / D row-half

    // Stage-1 matrix: blockdiag(W8, W8)  (two DFT-8 per 16-column)
    v2f Ar1[4], Ai1[4], AiN1[4];
#pragma unroll
    for (int kc = 0; kc < 4; ++kc) {
        int k0 = 4 * kc + ksel, k1 = k0 + 1;
        float c0 = 0.f, s0 = 0.f, c1 = 0.f, s1 = 0.f;
        if ((m >> 3) == (k0 >> 3)) { float a = -0.78539816339f * (float)(((m & 7) * (k0 & 7)) & 7); __sincosf(a, &s0, &c0); }
        if ((m >> 3) == (k1 >> 3)) { float a = -0.78539816339f * (float)(((m & 7) * (k1 & 7)) & 7); __sincosf(a, &s1, &c1); }
        Ar1[kc] = (v2f){c0, c1}; Ai1[kc] = (v2f){s0, s1}; AiN1[kc] = (v2f){-s0, -s1};
    }
    // Stage-2/3 matrix: W16
    v2f Ar2[4], Ai2[4], AiN2[4];
#pragma unroll
    for (int kc = 0; kc < 4; ++kc) {
        int k0 = 4 * kc + ksel, k1 = k0 + 1;
        float c0, s0, c1, s1;
        __sincosf(-0.39269908170f * (float)((m * k0) & 15), &s0, &c0);
        __sincosf(-0.39269908170f * (float)((m * k1) & 15), &s1, &c1);
        Ar2[kc] = (v2f){c0, c1}; Ai2[kc] = (v2f){s0, s1}; AiN2[kc] = (v2f){-s0, -s1};
    }

    // ---- stage 1: 256 DFT-8 over m2 (stride 256), in-place, tile = wave
    {
        v2f Br[4], Bi[4];
        const int p = (wave << 4) + col;          // pair of DFT-8 batches
#pragma unroll
        for (int kc = 0; kc < 4; ++kc) {
            int r0 = 4 * kc + ksel, r1 = r0 + 1;
            int a0 = (2 * p + (r0 >> 3)) + 256 * (r0 & 7);
            int a1 = (2 * p + (r1 >> 3)) + 256 * (r1 & 7);
            Br[kc] = (v2f){sRe[a0], sRe[a1]}; Bi[kc] = (v2f){sIm[a0], sIm[a1]};
        }
        v8f Dr = {}, Di = {};
        wmma_cmul(Ar1, Ai1, AiN1, Br, Bi, Dr, Di);
#pragma unroll
        for (int v = 0; v < 8; ++v) {
            int r = v + 8 * hi;
            int d = (2 * p + (r >> 3)) + 256 * (r & 7);
            sRe[d] = Dr[v]; sIm[d] = Di[v];
        }
    }
    __syncthreads();

    // ---- twiddle 1: exp(-2i*pi*m1*k3/128)
#pragma unroll
    for (int q = 0; q < 8; ++q) {
        int e  = tid + 256 * q;
        int m1 = (e >> 4) & 15, k3 = e >> 8;
        float s, c;
        __sincosf(-0.049087385213f * (float)((m1 * k3) & 127), &s, &c);
        float re = sRe[e], im = sIm[e];
        sRe[e] = re * c - im * s; sIm[e] = re * s + im * c;
    }
    __syncthreads();

    // ---- stage 2: 128 DFT-16 over m1 (stride 16), s -> t
    {
        v2f Br[4], Bi[4];
#pragma unroll
        for (int kc = 0; kc < 4; ++kc) {
            int r0 = 4 * kc + ksel, r1 = r0 + 1;
            int a0 = col + 16 * r0 + 256 * wave;
            int a1 = col + 16 * r1 + 256 * wave;
            Br[kc] = (v2f){sRe[a0], sRe[a1]}; Bi[kc] = (v2f){sIm[a0], sIm[a1]};
        }
        v8f Dr = {}, Di = {};
        wmma_cmul(Ar2, Ai2, AiN2, Br, Bi, Dr, Di);
#pragma unroll
        for (int v = 0; v < 8; ++v) {
            int r = v + 8 * hi;
            int d = col + 16 * wave + 128 * r;
            tRe[d] = Dr[v]; tIm[d] = Di[v];
        }
    }
    __syncthreads();

    // ---- twiddle 2: exp(-2i*pi*n1*k2/2048)
#pragma unroll
    for (int q = 0; q < 8; ++q) {
        int e  = tid + 256 * q;
        int n1 = e & 15, k2 = e >> 4;
        float s, c;
        __sincosf(-0.0030679615758f * (float)((n1 * k2) & 2047), &s, &c);
        float re = tRe[e], im = tIm[e];
        tRe[e] = re * c - im * s; tIm[e] = re * s + im * c;
    }
    __syncthreads();

    // ---- stage 3: 128 DFT-16 over n1 (stride 1), t -> s (natural order out)
    {
        v2f Br[4], Bi[4];
        const int k2 = 16 * wave + col;
#pragma unroll
        for (int kc = 0; kc < 4; ++kc) {
            int r0 = 4 * kc + ksel, r1 = r0 + 1;
            int a0 = r0 + 16 * k2;
            int a1 = r1 + 16 * k2;
            Br[kc] = (v2f){tRe[a0], tRe[a1]}; Bi[kc] = (v2f){tIm[a0], tIm[a1]};
        }
        v8f Dr = {}, Di = {};
        wmma_cmul(Ar2, Ai2, AiN2, Br, Bi, Dr, Di);
#pragma unroll
        for (int v = 0; v < 8; ++v) {
            int r = v + 8 * hi;
            int d = k2 + 128 * r;
            sRe[d] = Dr[v]; sIm[d] = Di[v];
        }
    }
    __syncthreads();

#pragma unroll
    for (int q = 0; q < 8; ++q) {
        int e = tid + 256 * q;
        row[e] = (float2){sRe[e], sIm[e]};
    }
}

// ----------------------------------------- in-place tiled square transpose
__global__ void transpose_inplace(float2* __restrict__ buf) {
    __shared__ float2 tA[32][33], tB[32][33];
    int b = blockIdx.x;
    int c = b / 2080, p = b % 2080;
    int bi = 0, rem = 64;
    while (p >= rem) { p -= rem; --rem; ++bi; }
    int bj = bi + p;

    int tx = threadIdx.x & 31, ty = threadIdx.x >> 5;
    float2* cb = buf + (size_t)c * GSZ * GSZ;
    bool diag = (bi == bj);
#pragma unroll
    for (int k = 0; k < 4; ++k) {
        int r = ty + 8 * k;
        tA[r][tx] = cb[(size_t)(bi * 32 + r) * GSZ + (bj * 32 + tx)];
        if (!diag) tB[r][tx] = cb[(size_t)(bj * 32 + r) * GSZ + (bi * 32 + tx)];
    }
    __syncthreads();
#pragma unroll
    for (int k = 0; k < 4; ++k) {
        int r = ty + 8 * k;
        cb[(size_t)(bj * 32 + r) * GSZ + (bi * 32 + tx)] = tA[tx][r];
        if (!diag) cb[(size_t)(bi * 32 + r) * GSZ + (bj * 32 + tx)] = tB[tx][r];
    }
}

// -------------------------------------------------- degrid (6x6 KB gather)
// After pass2 on transposed data: buf[c][kx][ky] == F[c][ky][kx].
__global__ void degrid(const float* __restrict__ uu, const float* __restrict__ vv,
                       const float2* __restrict__ F, float* __restrict__ out) {
    int k = blockIdx.x * blockDim.x + threadIdx.x;
    if (k >= NVIS) return;
    float tx = uu[k] * UVSCL, ty = vv[k] * UVSCL;
    int fx = (int)floorf(tx), fy = (int)floorf(ty);
    float wx[6], wy[6];
    int ix[6], iy[6];
#pragma unroll
    for (int a = 0; a < 6; ++a) {
        int off = a - 2;   // offs = arange(6) - (J/2 - 1)
        wy[a] = kbw(ty - (float)(fy + off));
        wx[a] = kbw(tx - (float)(fx + off));
        iy[a] = (fy + off) & (GSZ - 1);
        ix[a] = (fx + off) & (GSZ - 1);
    }
    float accR[NCHAN] = {0, 0, 0, 0}, accI[NCHAN] = {0, 0, 0, 0};
#pragma unroll
    for (int b = 0; b < 6; ++b) {
        size_t colbase = (size_t)ix[b] * GSZ;
#pragma unroll
        for (int a = 0; a < 6; ++a) {
            float  w = wy[a] * wx[b];
            size_t o = colbase + iy[a];
#pragma unroll
            for (int c = 0; c < NCHAN; ++c) {
                float2 f = F[(size_t)c * GSZ * GSZ + o];
                accR[c] += f.x * w; accI[c] += f.y * w;
            }
        }
    }
#pragma unroll
    for (int c = 0; c < NCHAN; ++c) {
        size_t o = 2 * ((size_t)c * NVIS + k);
        out[o] = accR[c]; out[o + 1] = accI[c];
    }
}

// ------------------------------------------------------------------ launch
extern "C" void kernel_launch(void* const* d_in, const int* in_sizes, int n_in,
                              void* d_out, int out_size, void* d_ws, size_t ws_size,
                              hipStream_t stream) {
    const float* img = (const float*)d_in[0];
    const float* uu  = (const float*)d_in[1];
    const float* vv  = (const float*)d_in[2];
    float*  out  = (float*)d_out;
    float*  apod = (float*)d_ws;
    float2* buf  = (float2*)((char*)d_ws + 8192);  // 4 x 2048 x 2048 complex

    apod_kernel     <<<4, 256, 0, stream>>>(apod);
    fill_grid       <<<(NCHAN * GSZ * GSZ) / 256, 256, 0, stream>>>(img, apod, buf);
    fft_rows        <<<NCHAN * GSZ, 256, 0, stream>>>(buf);   // row FFTs
    transpose_inplace<<<NCHAN * 2080, 256, 0, stream>>>(buf);
    fft_rows        <<<NCHAN * GSZ, 256, 0, stream>>>(buf);   // column FFTs
    degrid          <<<(NVIS + 255) / 256, 256, 0, stream>>>(uu, vv, buf, out);
}